// DilatedReparamConv_56495999811955
// MI455X (gfx1250) — compile-verified
//
#include <hip/hip_runtime.h>
#include <cstdint>

#define NCH   256      // channels
#define IMW   64       // width
#define IMH   64       // height
#define TH    32       // output rows per block
#define HALO  6        // max halo over all branches
#define PW    81       // padded tile row stride (odd -> bank-conflict-free)
#define PH    (TH + 2*HALO)

__device__ __forceinline__ float silu_bn(float y, float s, float b) {
    float z = fmaf(y, s, b);
    // z * sigmoid(z) = z / (1 + exp(-z)); v_exp_f32 + v_rcp_f32
    return z * __builtin_amdgcn_rcpf(1.0f + __expf(-z));
}

template<int KS, int DIL>
__device__ __forceinline__ void conv_branch(const float* __restrict__ t,
                                            int ty, int xb,
                                            const float* __restrict__ w,
                                            float a[8]) {
    constexpr int H = DIL * (KS - 1) / 2;
    constexpr int E = (KS - 1) * DIL + 8;   // register window extent per row
    #pragma unroll
    for (int ky = 0; ky < KS; ++ky) {
        const float* rp = t + (ty + HALO - H + ky * DIL) * PW + (xb + HALO - H);
        float win[E];
        #pragma unroll
        for (int e = 0; e < E; ++e) win[e] = rp[e];
        #pragma unroll
        for (int kx = 0; kx < KS; ++kx) {
            float ww = w[ky * KS + kx];
            #pragma unroll
            for (int j = 0; j < 8; ++j)
                a[j] = fmaf(win[kx * DIL + j], ww, a[j]);
        }
    }
}

__global__ __launch_bounds__(256)
void dilated_reparam_fused(const float* __restrict__ x,
                           const float* __restrict__ id_bn,   // [4,C]
                           const float* __restrict__ w5,      // [C,1,5,5]
                           const float* __restrict__ w7,      // [C,1,7,7]
                           const float* __restrict__ w3a,
                           const float* __restrict__ w3b,
                           const float* __restrict__ w3c,
                           const float* __restrict__ bng,     // [5,C]
                           const float* __restrict__ bnb,
                           const float* __restrict__ bnm,
                           const float* __restrict__ bnv,
                           float* __restrict__ out) {
    __shared__ float tile[PH * PW];
    __shared__ float wsh[104];   // 25 + 49 + 9 + 9 + 9 = 101
    __shared__ float bnsh[12];   // 5 branch (scale,bias) pairs + identity pair

    const int tid = threadIdx.x;
    const int blk = blockIdx.x;
    const int y0  = (blk & 1) * TH;          // 2 slabs per (n,ch)
    const int chn = blk >> 1;                // n*C + ch
    const int ch  = chn & (NCH - 1);
    const float* src = x   + (size_t)chn * (IMW * IMH);
    float*       dst = out + (size_t)chn * (IMW * IMH);

    // ---- stage per-channel weights + folded BN constants (uniform per block)
    if (tid < 25)       wsh[tid] = w5 [ch * 25 + tid];
    else if (tid < 74)  wsh[tid] = w7 [ch * 49 + (tid - 25)];
    else if (tid < 83)  wsh[tid] = w3a[ch * 9  + (tid - 74)];
    else if (tid < 92)  wsh[tid] = w3b[ch * 9  + (tid - 83)];
    else if (tid < 101) wsh[tid] = w3c[ch * 9  + (tid - 92)];
    else if (tid < 106) {
        int i = tid - 101;
        float s = bng[i * NCH + ch] * rsqrtf(bnv[i * NCH + ch] + 1e-5f);
        bnsh[2 * i]     = s;
        bnsh[2 * i + 1] = fmaf(-bnm[i * NCH + ch], s, bnb[i * NCH + ch]);
    } else if (tid == 106) {
        float s = id_bn[0 * NCH + ch] * rsqrtf(id_bn[3 * NCH + ch] + 1e-5f);
        bnsh[10] = s;
        bnsh[11] = fmaf(-id_bn[2 * NCH + ch], s, id_bn[1 * NCH + ch]);
    }

    // ---- stage input tile: CDNA5 async global->LDS for interior, ds-store 0 halo
    for (int i = tid; i < PH * PW; i += 256) {
        int r  = i / PW;
        int c  = i - r * PW;
        int gy = y0 - HALO + r;
        int gx = c - HALO;
        if (gx >= 0 && gx < IMW && gy >= 0 && gy < IMH) {
            const float* gp = src + gy * IMW + gx;
            unsigned lo = (unsigned)(uintptr_t)(&tile[i]);   // LDS byte offset
            asm volatile("global_load_async_to_lds_b32 %0, %1, off"
                         :: "v"(lo), "v"(gp) : "memory");
        } else {
            tile[i] = 0.0f;   // disjoint addresses from async targets
        }
    }
#if __has_builtin(__builtin_amdgcn_s_wait_asynccnt)
    __builtin_amdgcn_s_wait_asynccnt(0);
#else
    asm volatile("s_wait_asynccnt 0" ::: "memory");
#endif
    __syncthreads();

    const int ty = tid >> 3;        // 0..31
    const int xb = (tid & 7) * 8;   // 8 outputs per thread

    // identity branch
    float o[8];
    {
        const float id_s = bnsh[10], id_b = bnsh[11];
        const float* cp = &tile[(ty + HALO) * PW + (xb + HALO)];
        #pragma unroll
        for (int j = 0; j < 8; ++j) o[j] = fmaf(cp[j], id_s, id_b);
    }
    // branch 0: 5x5 dil 1
    {
        float a[8] = {0,0,0,0,0,0,0,0};
        conv_branch<5, 1>(tile, ty, xb, wsh + 0, a);
        float s = bnsh[0], b = bnsh[1];
        #pragma unroll
        for (int j = 0; j < 8; ++j) o[j] += silu_bn(a[j], s, b);
    }
    // branch 1: 7x7 dil 2
    {
        float a[8] = {0,0,0,0,0,0,0,0};
        conv_branch<7, 2>(tile, ty, xb, wsh + 25, a);
        float s = bnsh[2], b = bnsh[3];
        #pragma unroll
        for (int j = 0; j < 8; ++j) o[j] += silu_bn(a[j], s, b);
    }
    // branch 2: 3x3 dil 3
    {
        float a[8] = {0,0,0,0,0,0,0,0};
        conv_branch<3, 3>(tile, ty, xb, wsh + 74, a);
        float s = bnsh[4], b = bnsh[5];
        #pragma unroll
        for (int j = 0; j < 8; ++j) o[j] += silu_bn(a[j], s, b);
    }
    // branch 3: 3x3 dil 4
    {
        float a[8] = {0,0,0,0,0,0,0,0};
        conv_branch<3, 4>(tile, ty, xb, wsh + 83, a);
        float s = bnsh[6], b = bnsh[7];
        #pragma unroll
        for (int j = 0; j < 8; ++j) o[j] += silu_bn(a[j], s, b);
    }
    // branch 4: 3x3 dil 5
    {
        float a[8] = {0,0,0,0,0,0,0,0};
        conv_branch<3, 5>(tile, ty, xb, wsh + 92, a);
        float s = bnsh[8], b = bnsh[9];
        #pragma unroll
        for (int j = 0; j < 8; ++j) o[j] += silu_bn(a[j], s, b);
    }

    float* op = dst + (y0 + ty) * IMW + xb;      // 32-byte aligned
    *(float4*)(op)     = make_float4(o[0], o[1], o[2], o[3]);
    *(float4*)(op + 4) = make_float4(o[4], o[5], o[6], o[7]);
}

extern "C" void kernel_launch(void* const* d_in, const int* in_sizes, int n_in,
                              void* d_out, int out_size, void* d_ws, size_t ws_size,
                              hipStream_t stream) {
    (void)in_sizes; (void)n_in; (void)out_size; (void)d_ws; (void)ws_size;
    const float* x    = (const float*)d_in[0];
    const float* idbn = (const float*)d_in[1];
    const float* w5   = (const float*)d_in[2];
    const float* w7   = (const float*)d_in[3];
    const float* w3a  = (const float*)d_in[4];
    const float* w3b  = (const float*)d_in[5];
    const float* w3c  = (const float*)d_in[6];
    const float* bng  = (const float*)d_in[7];
    const float* bnb  = (const float*)d_in[8];
    const float* bnm  = (const float*)d_in[9];
    const float* bnv  = (const float*)d_in[10];
    float* out = (float*)d_out;

    dim3 grid(32 * NCH * (IMH / TH));   // 16384 blocks
    dim3 block(256);
    dilated_reparam_fused<<<grid, block, 0, stream>>>(
        x, idbn, w5, w7, w3a, w3b, w3c, bng, bnb, bnm, bnv, out);
}